// NF4DequantEmbedding_91010357002434
// MI455X (gfx1250) — compile-verified
//
#include <hip/hip_runtime.h>
#include <hip/hip_bf16.h>
#include <hip/hip_fp16.h>

// NF4 dequant embedding gather for MI455X (gfx1250, wave32).
// Bandwidth-bound: ~344 MB total traffic -> ~15us floor @ 23.3 TB/s.
// Strategy: 256-entry LDS LUT (byte -> float2 of the two nibble codebook
// values, ds_load_b64), 128-bit qweight loads, packed f32 FMA
// (v_pk_fma_f32), native packed f32->bf16 convert, non-temporal 128-bit
// stores (output stream > L2; keep L2 for the reused qweight table).

#define NF4_VOCAB  50257
#define NF4_D      4096
#define NF4_QBYTES (NF4_D / 2)   // 2048 packed bytes per row
#define NF4_GROUPS 64            // 64 groups of 64 elements
#define TPR        128           // threads per token row (4 waves)
#define RPB        2             // rows per block
#define BLOCK      (TPR * RPB)   // 256 threads = 8 wave32

// clang-native vector types (HIP_vector_type structs are rejected by
// __builtin_nontemporal_store)
typedef unsigned int v4u  __attribute__((ext_vector_type(4)));
typedef float        v2f  __attribute__((ext_vector_type(2)));
typedef __bf16       v2bf __attribute__((ext_vector_type(2)));

// fma two f32 lanes (v_pk_fma_f32 candidate) then pack both to bf16
// (v_cvt_pk_bf16_f32 / v_fma_mix*_bf16 candidate via backend matching).
static __device__ __forceinline__ uint32_t dequant_pair(v2f p, v2f sc2, v2f mn2) {
    const v2f r = __builtin_elementwise_fma(p, sc2, mn2);
    v2bf bb;
    bb.x = (__bf16)r.x;   // element 2k   (low nibble)
    bb.y = (__bf16)r.y;   // element 2k+1 (high nibble)
    return __builtin_bit_cast(uint32_t, bb);
}

__global__ __launch_bounds__(BLOCK) void nf4_embed_kernel(
    const int* __restrict__ input_ids,        // [T]
    const unsigned char* __restrict__ qweight,// [VOCAB, 2048] packed nibbles
    const __half* __restrict__ scale,         // [VOCAB, 64]
    const __half* __restrict__ mean,          // [VOCAB, 64]
    const __half* __restrict__ codebook,      // [16]
    const int* __restrict__ vocab_start,      // [1]
    __hip_bfloat16* __restrict__ out,         // [T, 4096]
    int num_tokens)
{
    __shared__ v2f lut[256];                  // byte -> (cb[lo], cb[hi])

    const int tid = threadIdx.x;
    {   // build dequant pair table once per block (BLOCK == 256)
        v2f p;
        p.x = __half2float(codebook[tid & 15]);
        p.y = __half2float(codebook[tid >> 4]);
        lut[tid] = p;
    }
    __syncthreads();

    const int row_in_blk = tid >> 7;              // 0..RPB-1
    const int t          = tid & (TPR - 1);       // lane-in-row 0..127
    const int token      = blockIdx.x * RPB + row_in_blk;
    if (token >= num_tokens) return;

    const int vs    = vocab_start[0];
    const int local = input_ids[token] - vs;
    const bool valid = (local >= 0) && (local < NF4_VOCAB);

    // each thread owns 32 consecutive output elements: [t*32, t*32+32)
    v4u* __restrict__ dst =
        (v4u*)(out + (size_t)token * NF4_D + (size_t)t * 32);

    if (!valid) {
        v4u z = (v4u)(0u);
        __builtin_nontemporal_store(z, dst + 0);
        __builtin_nontemporal_store(z, dst + 1);
        __builtin_nontemporal_store(z, dst + 2);
        __builtin_nontemporal_store(z, dst + 3);
        return;
    }

    const size_t row = (size_t)local;
    // 16 packed bytes = 32 nibbles = 32 outputs, one b128 load
    const v4u qw = ((const v4u*)(qweight + row * NF4_QBYTES))[t];
    // 32 elements stay inside one 64-wide group: group = t/2
    const int g = t >> 1;
    const float sc = __half2float(scale[row * NF4_GROUPS + g]);
    const float mn = __half2float(mean [row * NF4_GROUPS + g]);
    const v2f sc2 = { sc, sc };
    const v2f mn2 = { mn, mn };

    const uint32_t w[4] = { qw.x, qw.y, qw.z, qw.w };
    uint32_t o[16];

    #pragma unroll
    for (int i = 0; i < 4; ++i) {
        const uint32_t word = w[i];
        #pragma unroll
        for (int b = 0; b < 4; ++b) {
            const uint32_t byte = (word >> (8 * b)) & 0xFFu;
            const v2f p = lut[byte];              // ds_load_b64: both nibbles
            o[i * 4 + b] = dequant_pair(p, sc2, mn2);
        }
    }

    #pragma unroll
    for (int i = 0; i < 4; ++i) {
        v4u s;
        s.x = o[i * 4 + 0];
        s.y = o[i * 4 + 1];
        s.z = o[i * 4 + 2];
        s.w = o[i * 4 + 3];
        __builtin_nontemporal_store(s, dst + i);  // NT: output is write-once
    }
}

extern "C" void kernel_launch(void* const* d_in, const int* in_sizes, int n_in,
                              void* d_out, int out_size, void* d_ws, size_t ws_size,
                              hipStream_t stream) {
    // setup_inputs order:
    // 0: input_ids (int32, [4,8192])   1: qweight (uint8, [VOCAB, 2048])
    // 2: scale (fp16, [VOCAB,64])      3: mean (fp16, [VOCAB,64])
    // 4: codebook (fp16, [16])         5: group_size (int, [1])
    // 6: vocab_start_id (int, [1])
    const int*            input_ids   = (const int*)d_in[0];
    const unsigned char*  qweight     = (const unsigned char*)d_in[1];
    const __half*         scale       = (const __half*)d_in[2];
    const __half*         mean        = (const __half*)d_in[3];
    const __half*         codebook    = (const __half*)d_in[4];
    const int*            vocab_start = (const int*)d_in[6];
    __hip_bfloat16*       out         = (__hip_bfloat16*)d_out;

    const int n_tokens = in_sizes[0];
    dim3 grid((n_tokens + RPB - 1) / RPB);
    nf4_embed_kernel<<<grid, BLOCK, 0, stream>>>(
        input_ids, qweight, scale, mean, codebook, vocab_start, out, n_tokens);
}